// LowLevelInferenceEngine_42726334660997
// MI455X (gfx1250) — compile-verified
//
#include <hip/hip_runtime.h>

#define N_SYM    128
#define BATCH    131072
#define LEVERAGE 30.0f
#define PFDIST   8        // prefetch distance in symbols; must be >= unroll factor

// One thread per batch column. Each thread runs the 128-symbol recurrence,
// carrying unused_margin in a register. All global traffic is lane-coalesced
// (consecutive b across lanes -> one 128B line per wave per load) and marked
// non-temporal (stream-once, working set > 192MB L2). Prefetch PFDIST symbols
// (4MB/array) ahead via gfx1250 global_prefetch_b8, branch-free.
//
// Addressing: 32-bit unsigned element offsets (max < 2^24 elements, 2^26
// bytes) so loads lower to GVS form (SGPR64 base + VGPR32 offset,
// scale_offset) with per-symbol deltas folded into the 24-bit immediate —
// one 32-bit index add per unrolled body instead of 64-bit per-lane math.
__global__ __launch_bounds__(256)
void llie_pos_scan_kernel(const float* __restrict__ fr,   // fractions      [N_SYM,BATCH]
                          const float* __restrict__ ex,   // exec_samples   [N_SYM,BATCH]
                          const float* __restrict__ ps,   // pos_sizes      [N_SYM,BATCH]
                          const float* __restrict__ mr,   // margin_rates   [N_SYM,BATCH]
                          const float* __restrict__ um0,  // unused_margin  [BATCH]
                          const float* __restrict__ mn,   // msm_min  [N_SYM]
                          const float* __restrict__ st,   // msm_step [N_SYM]
                          const float* __restrict__ mx,   // msm_max  [N_SYM]
                          float* __restrict__ out_pos,    // [N_SYM,BATCH]
                          float* __restrict__ out_um)     // [BATCH]
{
    const unsigned b = blockIdx.x * blockDim.x + threadIdx.x;

    float um = um0[b];

    unsigned idx = b;                  // 32-bit element offset: s*BATCH + b
    #pragma unroll 4
    for (int s = 0; s < N_SYM; ++s, idx += BATCH) {
        // Branch-free speculative prefetch PFDIST symbols ahead
        // (global_prefetch_b8). In the last PFDIST iterations the offset
        // clamps to 0: re-prefetching the line we are about to demand-load
        // is harmless and keeps the unrolled body straight-line.
        const unsigned pofs = (s + PFDIST < N_SYM) ? (unsigned)(PFDIST * BATCH) : 0u;
        const unsigned pidx = idx + pofs;
        __builtin_prefetch(fr + pidx, 0, 0);
        __builtin_prefetch(ex + pidx, 0, 0);
        __builtin_prefetch(ps + pidx, 0, 0);
        __builtin_prefetch(mr + pidx, 0, 0);

        // Stream-once inputs: non-temporal loads (TH_LOAD_NT), no L2 thrash.
        const float f = __builtin_nontemporal_load(fr + idx);
        const float e = __builtin_nontemporal_load(ex + idx);
        const float p = __builtin_nontemporal_load(ps + idx);
        const float m = __builtin_nontemporal_load(mr + idx);

        // Loop-uniform per-symbol constants -> scalar (constant-cache) loads.
        const float s_mn = mn[s];
        const float s_st = st[s];
        const float s_mx = mx[s];

        const float t   = m * LEVERAGE;          // mr * LEVERAGE
        const float us  = um * t;                // unused_size
        const float mps = fabsf(p) + us;         // max_pos_size
        // min(where(pos*frac > 0, unused_size, max_pos_size), 0)
        const float adj = fminf((p * f > 0.0f) ? us : mps, 0.0f);
        const float nps = f * (mps - adj);       // new_pos_size (pre-quantization)

        float a = fabsf(nps);
        const float sg = (nps > 0.0f) ? 1.0f : ((nps < 0.0f) ? -1.0f : 0.0f);
        a = floorf(a / s_st) * s_st;             // quantize to step (IEEE div)
        a = (a < s_mn) ? 0.0f : a;               // below min -> 0
        a = fminf(a, s_mx);                      // clamp to max

        const float o = e * (sg * a) + (1.0f - e) * p;
        __builtin_nontemporal_store(o, out_pos + idx);

        // carried recurrence (precise IEEE divide: error must not compound)
        um = (mps - a) / t;
    }

    out_um[b] = um;
}

extern "C" void kernel_launch(void* const* d_in, const int* in_sizes, int n_in,
                              void* d_out, int out_size, void* d_ws, size_t ws_size,
                              hipStream_t stream) {
    const float* fr  = (const float*)d_in[0];  // fractions
    const float* ex  = (const float*)d_in[1];  // exec_samples
    const float* ps  = (const float*)d_in[2];  // pos_sizes
    const float* mr  = (const float*)d_in[3];  // margin_rates
    const float* um0 = (const float*)d_in[4];  // unused_margin
    const float* mn  = (const float*)d_in[5];  // msm_min
    const float* st  = (const float*)d_in[6];  // msm_step
    const float* mx  = (const float*)d_in[7];  // msm_max

    float* out_pos = (float*)d_out;                       // [N_SYM*BATCH]
    float* out_um  = out_pos + (size_t)N_SYM * BATCH;     // [BATCH]

    dim3 block(256);                 // 8 wave32 per block
    dim3 grid(BATCH / 256);          // 512 blocks -> 4096 wave32 in flight

    hipLaunchKernelGGL(llie_pos_scan_kernel, grid, block, 0, stream,
                       fr, ex, ps, mr, um0, mn, st, mx, out_pos, out_um);
}